// SoilModel_76373108457545
// MI455X (gfx1250) — compile-verified
//
#include <hip/hip_runtime.h>
#include <hip/hip_bf16.h>
#include <cstdint>

static constexpr int NT = 1000;
static constexpr int NS = 256;
static constexpr int NH = 128;

__device__ __forceinline__ float sigm(float x) { return 1.0f / (1.0f + expf(-x)); }

// -------- Pass A: forcing precompute: {rain, snow, melt-temp, PET} per (t,ns) --------
// Hoists acos / temperature branching out of the 1000-step serial scan.
__global__ __launch_bounds__(256) void forcing_kernel(
    const float* __restrict__ P, const float* __restrict__ T1,
    const float* __restrict__ T2, const float* __restrict__ E,
    float4* __restrict__ F) {
  int idx = blockIdx.x * 256 + threadIdx.x;
  if (idx >= NT * NS) return;
  float p = P[idx], t1 = T1[idx], t2 = T2[idx], e = E[idx];
  float arg = (t1 < 0.0f && t2 > 0.0f) ? (t1 + t2) / (t2 - t1) : 0.0f;
  float rP = 1.0f - acosf(arg) / 3.1415f;   // reference uses PI = 3.1415
  if (t1 >= 0.0f) rP = 1.0f;
  if (t2 <= 0.0f) rP = 0.0f;                // T2<=0 takes precedence, as in reference
  float mt = fmaxf(0.5f * (t1 + t2), 0.0f);
  F[idx] = make_float4(rP * p, (1.0f - rP) * p, mt, e);
}

// Issue one async global->LDS B128 per lane: lane <-> step within the 32-step chunk.
// CDNA5 async copy engine (ASYNCcnt); data lands in this wave's private LDS slab.
__device__ __forceinline__ void async_stage(const float4* __restrict__ F, int ns, int t0,
                                            unsigned ldsBase, int lane) {
  int t = t0 + lane;
  t = (t < NT) ? t : (NT - 1);                       // clamp tail chunk (harmless reload)
  const float4* src = F + ((size_t)t * NS + ns);
  unsigned lds = ldsBase + (unsigned)lane * 16u;     // LDS byte addr = low 32 bits of generic ptr
  unsigned long long ga = (unsigned long long)(uintptr_t)src;
  asm volatile("global_load_async_to_lds_b128 %0, %1, off offset:0"
               :: "v"(lds), "v"(ga) : "memory");
}

// -------- Main scan: 1 thread = 1 (ns,nh) cell, 1000 sequential steps in registers --------
// Block = 256 threads = 2 ns rows x 128 nh; 128 blocks -> 1024 wave32s machine-wide.
__global__ __launch_bounds__(256) void soil_scan(
    const float4* __restrict__ F,
    const float* __restrict__ wo, const float* __restrict__ b,
    const float* __restrict__ sw, const float* __restrict__ wk1,
    const float* __restrict__ wk2, const float* __restrict__ we1,
    const float* __restrict__ we2, const float* __restrict__ wl,
    const float* __restrict__ gwk,
    float* __restrict__ Q, float* __restrict__ S, float* __restrict__ H1,
    float* __restrict__ H2, float* __restrict__ Qs) {
  __shared__ float4 fb[8][2][32];        // [wave][buffer][step-in-chunk] forcing, 8KB
  __shared__ float  part[8][2][32][4];   // [wave][chunk-parity][step][r1,r2,rg,pad], 8KB

  const int tid  = threadIdx.x;
  const int nsl  = tid >> 7;             // 0/1: which ns row in block
  const int nh   = tid & (NH - 1);
  const int w    = tid >> 5;             // wave 0..7 (0..3 -> ns row 0, 4..7 -> ns row 1)
  const int lane = tid & 31;
  const int ns   = blockIdx.x * 2 + nsl;

  // softmax denominator and the constant bias term  Cb = sum(relu(b)*a)  (fold once)
  float D = 0.0f, Csum = 0.0f;
  for (int i = 0; i < NH; ++i) {
    float e = expf(wo[i]);
    D += e;
    Csum += e * fmaxf(b[i], 0.0f);
  }
  const float a   = expf(wo[nh]) / D;
  const float Cb  = Csum / D;
  const float smg = expf(sw[nh]) + 1.0f;
  const float sk1 = sigm(wk1[nh]);
  const float sk2 = sigm(wk2[nh]);       // reference uses wk2 twice; wk3 is dead
  const float se1 = sigm(we1[nh]);
  const float se2 = sigm(we2[nh]);
  const float Lh  = expf(wl[nh]);
  const float sg  = sigm(gwk[nh]);

  float s = 0.0f, h1 = 0.0f, h2 = 0.0f, h3 = 0.0f;

  const size_t cell       = (size_t)ns * NH + nh;
  const size_t stepstride = (size_t)NS * NH;

  const unsigned fb0 = (unsigned)(unsigned long long)(uintptr_t)&fb[w][0][0];
  const unsigned fb1 = (unsigned)(unsigned long long)(uintptr_t)&fb[w][1][0];

  // prime the double buffer (ASYNCcnt = 2 outstanding per wave)
  async_stage(F, ns, 0, fb0, lane);
  async_stage(F, ns, 32, fb1, lane);

  int t = 0;
  for (int c = 0; c < 32; ++c) {                     // 32 chunks x 32 steps (last = 8)
    if (c == 31) asm volatile("s_wait_asynccnt 0x0" ::: "memory");
    else         asm volatile("s_wait_asynccnt 0x1" ::: "memory");
    const int nst = (NT - c * 32 < 32) ? (NT - c * 32) : 32;
    const int par = c & 1;
    const float4* fbp = &fb[w][par][0];

    for (int si = 0; si < nst; ++si, ++t) {
      const float4 f = fbp[si];          // {rain, snow, melt, PET} : ds_load_b128, uniform
      // ---- SnowNet ----
      s += f.y;
      float sm  = f.z * smg;
      float m   = fminf(sm, s);          // NM == 0
      s -= m;
      float x   = f.x + m;
      // ---- SoilBucket ----
      float e1  = f.w * se1;
      float e2  = f.w * se2;
      float h2x = h2 + x;
      float h1n = fmaxf(h1 + h2x - Lh, 0.0f);
      float q1  = h1n * sk1;
      float h2n = fminf(h1n + h2x, Lh);
      float q2  = h2n * sk2;             // q3 == q2 (wk2 reused)
      float h1f = fmaxf(h1n - q1 - e1, 0.0f);
      float h2f = fmaxf(h2n - q2 - q2 - e2, 0.0f);
      // ---- LinearBucket (GW) ----
      float qg  = (q2 + h3) * sg;
      h3 = h3 - qg + q2;
      h1 = h1f; h2 = h2f;

      // streaming state history: non-temporal, fully coalesced (32 lanes = 128B lines)
      const size_t off = (size_t)t * stepstride + cell;
      __builtin_nontemporal_store(s,   S  + off);
      __builtin_nontemporal_store(h1f, H1 + off);
      __builtin_nontemporal_store(h2f, H2 + off);

      // per-step weighted reductions over this wave's 32 nh lanes
      float r1 = q1 * a, r2 = q2 * a, rg = qg * a;
#pragma unroll
      for (int mk = 16; mk >= 1; mk >>= 1) {
        r1 += __shfl_xor(r1, mk, 32);
        r2 += __shfl_xor(r2, mk, 32);
        rg += __shfl_xor(rg, mk, 32);
      }
      if (lane == 0) {                   // wave-private slab: no sync needed here
        float4 v = make_float4(r1, r2, rg, 0.0f);
        *(float4*)&part[w][par][si][0] = v;
      }
    }

    // kick off the async refill for chunk c+2 before the barrier (overlaps finisher + next chunk)
    if (c + 2 < 32) async_stage(F, ns, (c + 2) * 32, par ? fb1 : fb0, lane);

    __syncthreads();                     // ONE barrier per 32 steps (parity-buffered part[])

    // finishers: wave 0 handles ns row 0, wave 4 handles ns row 1; lane <-> step in chunk
    if ((w & 3) == 0 && lane < nst) {
      const int tt = c * 32 + lane;
      float s1 = 0.0f, s2 = 0.0f, s3 = 0.0f;
#pragma unroll
      for (int wv = 0; wv < 4; ++wv) {
        const float4 v = *(const float4*)&part[(w & 4) + wv][par][lane][0];
        s1 += v.x; s2 += v.y; s3 += v.z;
      }
      Q[(size_t)tt * NS + ns] = s1 + s2 + s3 + Cb;
      float* qp = Qs + ((size_t)tt * NS + ns) * 3;
      qp[0] = s1; qp[1] = s2; qp[2] = s2;   // Qs col2 == col1 in reference
    }
  }
}

extern "C" void kernel_launch(void* const* d_in, const int* in_sizes, int n_in,
                              void* d_out, int out_size, void* d_ws, size_t ws_size,
                              hipStream_t stream) {
  const float* P   = (const float*)d_in[0];
  const float* T1  = (const float*)d_in[1];
  const float* T2  = (const float*)d_in[2];
  const float* E   = (const float*)d_in[3];
  const float* wo  = (const float*)d_in[4];
  const float* b   = (const float*)d_in[5];
  const float* sw  = (const float*)d_in[6];
  const float* wk1 = (const float*)d_in[7];
  const float* wk2 = (const float*)d_in[8];
  // d_in[9] = wk3: declared but unused by the reference math
  const float* we1 = (const float*)d_in[10];
  const float* we2 = (const float*)d_in[11];
  const float* wl  = (const float*)d_in[12];
  const float* gwk = (const float*)d_in[13];

  // d_out layout: Q [NT,NS] | S [NT,NS,NH] | H1 | H2 | Qs [NT,NS,3]
  float* Q  = (float*)d_out;
  float* S  = Q  + (size_t)NT * NS;
  float* H1 = S  + (size_t)NT * NS * NH;
  float* H2 = H1 + (size_t)NT * NS * NH;
  float* Qs = H2 + (size_t)NT * NS * NH;

  float4* F = (float4*)d_ws;             // NT*NS*16B = 4 MB scratch

  forcing_kernel<<<(NT * NS + 255) / 256, 256, 0, stream>>>(P, T1, T2, E, F);
  soil_scan<<<NS / 2, 256, 0, stream>>>(F, wo, b, sw, wk1, wk2, we1, we2, wl,
                                        gwk, Q, S, H1, H2, Qs);
}